// seq2seq_43439299231873
// MI455X (gfx1250) — compile-verified
//
#include <hip/hip_runtime.h>
#include <math.h>

typedef float v2f __attribute__((ext_vector_type(2)));
typedef float v8f __attribute__((ext_vector_type(8)));

#define HID 1024
#define VOC 50257
#define SEQ 2048

// ---------------------------------------------------------------------------
// One wave: y[row_base..row_base+15] = sum_{k in [k0,k0+KC)} W[row,k]*xs[k]
// via V_WMMA_F32_16X16X4_F32, two independent accumulator chains.
// A (16x4 f32): lanes 0-15 hold K=0,1 ; lanes 16-31 hold K=2,3.
// B (4x16 f32): x broadcast into all 16 columns (VGPR p = rows K=p,p+2).
// D (16x16 f32): columns identical; lane 0 holds rows 0-7 in acc[0..7],
// lane 16 holds rows 8-15.  KC is compile-time, multiple of 8.
// k0 and row_base must be wave-uniform scalars (readfirstlane'd).
// ---------------------------------------------------------------------------
template <int LDA, int KC>
__device__ __forceinline__ void wave_gemv16_wmma(
    const float* __restrict__ W, const float* __restrict__ xs,
    int k0, int row_base, int rows,
    const float* __restrict__ bias, float* __restrict__ y)
{
    const int lane = threadIdx.x & 31;
    const int half = lane >> 4;
    const int lrow = lane & 15;
    int row = row_base + lrow;
    if (row >= rows) row = rows - 1;              // clamp (results discarded)
    const int kh = half * 2;
    const float* __restrict__ pW = W + (size_t)row * LDA + k0 + kh;
    const float* __restrict__ px = xs + k0 + kh;

    v8f acc0 = {}, acc1 = {};
#pragma unroll 4
    for (int it = 0; it < KC / 8; ++it) {
        v2f a0 = *(const v2f*)(pW);
        v2f a1 = *(const v2f*)(pW + 4);
        v2f b0 = *(const v2f*)(px);
        v2f b1 = *(const v2f*)(px + 4);
        acc0 = __builtin_amdgcn_wmma_f32_16x16x4_f32(
            false, a0, false, b0, (short)0, acc0, false, false);
        acc1 = __builtin_amdgcn_wmma_f32_16x16x4_f32(
            false, a1, false, b1, (short)0, acc1, false, false);
        pW += 8;
        px += 8;
    }
    acc0 = acc0 + acc1;

    if (lrow == 0) {                              // lanes 0 and 16: 8 rows each
        const int mb = half * 8;
#pragma unroll
        for (int p = 0; p < 8; ++p) {
            int r = row_base + mb + p;
            if (r < rows) y[r] = acc0[p] + (bias ? bias[r] : 0.0f);
        }
    }
}

// Dual partial GEMV with compile-time K sizes and K-split:
//   ya[kc*rows + r] = Wa[r, kc-chunk] . xa  (+ba if kc==0); yb likewise.
// One wave per (16-row tile, K-chunk). x vectors staged in LDS.
template <int KA, int KB, int KSPLIT>
__global__ void __launch_bounds__(256)
gates_wmma_kernel(const float* __restrict__ Wa, const float* __restrict__ xa,
                  const float* __restrict__ ba, float* __restrict__ ya,
                  const float* __restrict__ Wb, const float* __restrict__ xb,
                  const float* __restrict__ bb, float* __restrict__ yb,
                  int rows)
{
    __shared__ float xs[KA + KB + 1];
    for (int i = threadIdx.x; i < KA; i += blockDim.x) xs[i] = xa[i];
    if (KB > 0)
        for (int i = threadIdx.x; i < KB; i += blockDim.x) xs[KA + i] = xb[i];
    __syncthreads();

    // wave-uniform task id, forced into SGPRs so loop control stays scalar
    const int task =
        __builtin_amdgcn_readfirstlane(blockIdx.x * (blockDim.x >> 5) +
                                       (threadIdx.x >> 5));
    const int tiles = (rows + 15) >> 4;
    if (task >= tiles * KSPLIT) return;           // wave-uniform exit
    const int tile = task % tiles;
    const int kc   = task / tiles;
    const int row_base = tile << 4;

    constexpr int KCA = KA / KSPLIT;
    wave_gemv16_wmma<KA, KCA>(Wa, xs, kc * KCA, row_base, rows,
                              (kc == 0) ? ba : nullptr,
                              ya + (size_t)kc * rows);
    if (KB > 0) {
        constexpr int KCB = (KB > 0) ? KB / KSPLIT : 8;
        wave_gemv16_wmma<KB ? KB : 8, KCB>(Wb, xs + KA, kc * KCB, row_base,
                                           rows, (kc == 0) ? bb : nullptr,
                                           yb + (size_t)kc * rows);
    }
}

// x0 = [ emb[word], last_context ]
__global__ void prep_kernel(const float* __restrict__ emb,
                            const int* __restrict__ word,
                            const float* __restrict__ last_ctx,
                            float* __restrict__ x0)
{
    int j = blockIdx.x * blockDim.x + threadIdx.x;
    if (j < HID)          x0[j] = emb[(size_t)word[0] * HID + j];
    else if (j < 2 * HID) x0[j] = last_ctx[j - HID];
}

// Sum K-split partials, apply GRU gate math.
__global__ void gru_combine_kernel(const float* __restrict__ gi,
                                   const float* __restrict__ gh,
                                   const float* __restrict__ hprev,
                                   float* __restrict__ h_ws,
                                   float* __restrict__ h_out, int nsplit)
{
    int j = blockIdx.x * blockDim.x + threadIdx.x;
    if (j >= HID) return;
    float ir = 0, iz = 0, in = 0, hr = 0, hz = 0, hn = 0;
    for (int c = 0; c < nsplit; ++c) {
        const int b = c * 3 * HID;
        ir += gi[b + j];           hr += gh[b + j];
        iz += gi[b + j + HID];     hz += gh[b + j + HID];
        in += gi[b + j + 2 * HID]; hn += gh[b + j + 2 * HID];
    }
    float r = 1.0f / (1.0f + expf(-(ir + hr)));
    float z = 1.0f / (1.0f + expf(-(iz + hz)));
    float n = tanhf(in + r * hn);
    float h = (1.0f - z) * n + z * hprev[j];
    h_ws[j]  = h;
    h_out[j] = h;
}

// softmax over SEQ energies stored as nsplit partial arrays (single block)
__global__ void __launch_bounds__(1024)
softmax2048_kernel(const float* __restrict__ e, int nsplit,
                   float* __restrict__ a_ws, float* __restrict__ a_out)
{
    __shared__ float sd[1024];
    const int t = threadIdx.x;
    float m = -INFINITY;
    for (int i = t; i < SEQ; i += 1024) {
        float v = 0.0f;
        for (int c = 0; c < nsplit; ++c) v += e[c * SEQ + i];
        m = fmaxf(m, v);
    }
    sd[t] = m; __syncthreads();
    for (int s = 512; s > 0; s >>= 1) {
        if (t < s) sd[t] = fmaxf(sd[t], sd[t + s]);
        __syncthreads();
    }
    m = sd[0]; __syncthreads();
    float sum = 0.0f;
    for (int i = t; i < SEQ; i += 1024) {
        float v = 0.0f;
        for (int c = 0; c < nsplit; ++c) v += e[c * SEQ + i];
        sum += expf(v - m);
    }
    sd[t] = sum; __syncthreads();
    for (int s = 512; s > 0; s >>= 1) {
        if (t < s) sd[t] += sd[t + s];
        __syncthreads();
    }
    float inv = 1.0f / sd[0];
    for (int i = t; i < SEQ; i += 1024) {
        float v = 0.0f;
        for (int c = 0; c < nsplit; ++c) v += e[c * SEQ + i];
        v = expf(v - m) * inv;
        a_ws[i]  = v;
        a_out[i] = v;
    }
}

// context partial: 16 row-chunks x 4 col-blocks = 64 blocks
__global__ void context_partial_kernel(const float* __restrict__ enc,
                                       const float* __restrict__ attn,
                                       float* __restrict__ part)
{
    const int cb = blockIdx.x & 3;          // column block (256 cols)
    const int rc = blockIdx.x >> 2;         // row chunk (128 rows)
    const int j  = cb * 256 + threadIdx.x;
    float acc = 0.0f;
    const int i0 = rc * (SEQ / 16);
    for (int i = i0; i < i0 + SEQ / 16; ++i)
        acc += attn[i] * enc[(size_t)i * HID + j];
    part[(size_t)rc * HID + j] = acc;
}

__global__ void context_combine_kernel(const float* __restrict__ part,
                                       float* __restrict__ ctx_ws,
                                       float* __restrict__ ctx_out)
{
    int j = blockIdx.x * blockDim.x + threadIdx.x;
    if (j >= HID) return;
    float acc = 0.0f;
    for (int c = 0; c < 16; ++c) acc += part[(size_t)c * HID + j];
    ctx_ws[j]  = acc;
    ctx_out[j] = acc;
}

// ---- parallel log-softmax (online (max,sum) reduction) ----------------------
__device__ __forceinline__ void ols_merge(float& m, float& s, float m2, float s2)
{
    float M = fmaxf(m, m2);
    s = s * expf(m - M) + s2 * expf(m2 - M);
    m = M;
}

__global__ void __launch_bounds__(256)
lsm_partial_kernel(const float* __restrict__ l, int N,
                   float* __restrict__ pm, float* __restrict__ ps)
{
    __shared__ float sm[256], ss[256];
    const int g = blockIdx.x * 256 + threadIdx.x;
    const int stride = gridDim.x * 256;
    float m = -INFINITY, s = 0.0f;
    for (int i = g; i < N; i += stride) {
        float v = l[i];
        if (v > m) { s = s * expf(m - v) + 1.0f; m = v; }
        else       { s += expf(v - m); }
    }
    sm[threadIdx.x] = m; ss[threadIdx.x] = s; __syncthreads();
    for (int st = 128; st > 0; st >>= 1) {
        if (threadIdx.x < st) {
            float m1 = sm[threadIdx.x], s1 = ss[threadIdx.x];
            ols_merge(m1, s1, sm[threadIdx.x + st], ss[threadIdx.x + st]);
            sm[threadIdx.x] = m1; ss[threadIdx.x] = s1;
        }
        __syncthreads();
    }
    if (threadIdx.x == 0) { pm[blockIdx.x] = sm[0]; ps[blockIdx.x] = ss[0]; }
}

__global__ void __launch_bounds__(64)
lsm_final_kernel(const float* __restrict__ pm, const float* __restrict__ ps,
                 float* __restrict__ lse)
{
    __shared__ float sm[64], ss[64];
    const int t = threadIdx.x;
    sm[t] = pm[t]; ss[t] = ps[t]; __syncthreads();
    for (int st = 32; st > 0; st >>= 1) {
        if (t < st) {
            float m1 = sm[t], s1 = ss[t];
            ols_merge(m1, s1, sm[t + st], ss[t + st]);
            sm[t] = m1; ss[t] = s1;
        }
        __syncthreads();
    }
    if (t == 0) lse[0] = sm[0] + logf(ss[0]);
}

__global__ void lsm_write_kernel(const float* __restrict__ l,
                                 const float* __restrict__ lse,
                                 float* __restrict__ out, int N)
{
    int i = blockIdx.x * blockDim.x + threadIdx.x;
    if (i < N) out[i] = l[i] - lse[0];
}

// ---------------------------------------------------------------------------
extern "C" void kernel_launch(void* const* d_in, const int* in_sizes, int n_in,
                              void* d_out, int out_size, void* d_ws, size_t ws_size,
                              hipStream_t stream)
{
    (void)in_sizes; (void)n_in; (void)out_size; (void)ws_size;
    const int*   word   = (const int*)  d_in[0];
    const float* lctx   = (const float*)d_in[1];
    const float* lhid   = (const float*)d_in[2];   // (2,1,1024)
    const float* enc    = (const float*)d_in[3];   // (2048,1,1024)
    const float* emb    = (const float*)d_in[4];
    const float* W_ih0  = (const float*)d_in[5];
    const float* W_hh0  = (const float*)d_in[6];
    const float* b_ih0  = (const float*)d_in[7];
    const float* b_hh0  = (const float*)d_in[8];
    const float* W_ih1  = (const float*)d_in[9];
    const float* W_hh1  = (const float*)d_in[10];
    const float* b_ih1  = (const float*)d_in[11];
    const float* b_hh1  = (const float*)d_in[12];
    const float* W_out  = (const float*)d_in[13];
    const float* b_out  = (const float*)d_in[14];

    float* ws = (float*)d_ws;
    float* x0    = ws;                  // 2048
    float* gi    = ws + 2048;           // 4*3072 = 12288 (K-split partials)
    float* gh    = ws + 14336;          // 4*3072 = 12288
    float* h0    = ws + 26624;          // 1024
    float* h1    = ws + 27648;          // 1024 } contiguous [h1|ctx] = xcat
    float* ctx   = ws + 28672;          // 1024 }
    float* en    = ws + 29696;          // 4*2048 = 8192 (K-split partials)
    float* att   = ws + 37888;          // 2048
    float* cpart = ws + 39936;          // 16*1024 = 16384
    float* lg    = ws + 56320;          // 50257
    float* pm    = ws + 106577;         // 64
    float* psum  = ws + 106641;         // 64
    float* lse   = ws + 106705;         // 1

    float* out   = (float*)d_out;
    float* o_lp  = out;                 // log-probs   (50257)
    float* o_ctx = out + VOC;           // context     (1024)
    float* o_hid = out + VOC + HID;     // h0,h1       (2048)
    float* o_att = out + VOC + 3 * HID; // attn        (2048)

    // 1. x0 = [emb[word], last_context]
    prep_kernel<<<8, 256, 0, stream>>>(emb, word, lctx, x0);

    // 2. layer-0 gates (192 tiles x 4 K-chunks = 768 wave-tasks -> 96 blocks)
    gates_wmma_kernel<2 * HID, HID, 4><<<96, 256, 0, stream>>>(
        W_ih0, x0, b_ih0, gi, W_hh0, lhid, b_hh0, gh, 3 * HID);
    // 3. h0
    gru_combine_kernel<<<4, 256, 0, stream>>>(gi, gh, lhid, h0, o_hid, 4);

    // 4. layer-1 gates
    gates_wmma_kernel<HID, HID, 4><<<96, 256, 0, stream>>>(
        W_ih1, h0, b_ih1, gi, W_hh1, lhid + HID, b_hh1, gh, 3 * HID);
    // 5. h1
    gru_combine_kernel<<<4, 256, 0, stream>>>(gi, gh, lhid + HID, h1,
                                              o_hid + HID, 4);

    // 6. energies = enc . h1  (128 tiles x 4 K-chunks -> 64 blocks)
    gates_wmma_kernel<HID, 0, 4><<<64, 256, 0, stream>>>(
        enc, h1, nullptr, en, nullptr, nullptr, nullptr, nullptr, SEQ);
    // 7. attn = softmax(sum of energy partials)
    softmax2048_kernel<<<1, 1024, 0, stream>>>(en, 4, att, o_att);

    // 8. context = attn . enc  (64-block partial + combine)
    context_partial_kernel<<<64, 256, 0, stream>>>(enc, att, cpart);
    context_combine_kernel<<<4, 256, 0, stream>>>(cpart, ctx, o_ctx);

    // 9. logits = W_out . [h1|ctx] + b_out  (3142 tiles -> 393 blocks)
    gates_wmma_kernel<2 * HID, 0, 1><<<393, 256, 0, stream>>>(
        W_out, h1, b_out, lg, nullptr, nullptr, nullptr, nullptr, VOC);
    // 10. output = log_softmax(logits), parallel online reduction
    lsm_partial_kernel<<<64, 256, 0, stream>>>(lg, VOC, pm, psum);
    lsm_final_kernel<<<1, 64, 0, stream>>>(pm, psum, lse);
    lsm_write_kernel<<<197, 256, 0, stream>>>(lg, lse, o_lp, VOC);
}